// Transformer_14353780703602
// MI455X (gfx1250) — compile-verified
//
#include <hip/hip_runtime.h>
#include <hip/hip_bf16.h>

// ---------------------------------------------------------------------------
// MI455X (gfx1250) transformer forward. Memory-bound (1.34 GB fp32 weights,
// ~20 GFLOP -> ~57us HBM floor @ 23.3 TB/s): stream weights via the Tensor
// Data Mover into double-buffered LDS tiles, fp32 WMMA 16x16x4, split-K=8
// (2048 waves/GEMM) with fp32 atomic accumulation, epilogues fused into
// consumers.
// ---------------------------------------------------------------------------

typedef float v2f __attribute__((ext_vector_type(2)));
typedef float v8f __attribute__((ext_vector_type(8)));
typedef unsigned int v4u __attribute__((ext_vector_type(4)));
typedef int v4i __attribute__((ext_vector_type(4)));
typedef int v8i __attribute__((ext_vector_type(8)));

#define S_   30
#define D_   4096
#define H_   8
#define L_   4
#define DH_  512
#define MPAD 32
#define ROWBUF (MPAD * D_)        // 131072 floats = one padded [32,4096] buffer
#define WAVES   4                 // waves per GEMM workgroup (64 output cols)
#define KSPLIT  8                 // split-K factor
#define KT      32                // K-tile staged per TDM transfer
#define EPS_ 1e-5f
#define INV_SCALE (1.0f / 64.0f)

// LDS tile layout (floats). A tile padded: TDM pad_interval=32dw,pad_amount=2dw
// -> row stride 34 floats, kills ds bank conflicts on the strided A reads.
#define A_STRIDE     34
#define A_BUF_FLOATS (32 * A_STRIDE)          // 1088
#define B_BUF_FLOATS (KT * 64)                // 2048
#define SMEM_FLOATS  (2 * A_BUF_FLOATS + 2 * B_BUF_FLOATS)   // 6272 = 25088 B

// D# group1 dword0: data_size=4B (2<<16); A adds pad_enable|interval=32dw|amount=2dw
#define TDM_D0_PLAIN (2u << 16)
#define TDM_D0_PADA  ((2u << 16) | (1u << 20) | (4u << 22) | (1u << 25))

// --------------------------------------------------------------- WMMA -------
__device__ __forceinline__ v8f wmma4(v2f a, v2f b, v8f c) {
    return __builtin_amdgcn_wmma_f32_16x16x4_f32(false, a, false, b,
                                                 (short)0, c, false, false);
}

// Publish TDM-written LDS to the compiler: empty asm that "may write" through
// the escaped shared pointer. Forces real ds_load fragment reads and pins them
// after the tensorcnt-wait + barrier.
__device__ __forceinline__ void lds_publish(void* p) {
    asm volatile("" : : "v"(p) : "memory");
}

// ------------------------------------------------ TDM 2-D tile -> LDS -------
// D# bitfields per CDNA5 ISA. Groups 2/3 zeroed
// (2-D tensor -> hardware sees the 2-group descriptor form).
__device__ __forceinline__ void tdm_load_2d(unsigned ldsByte, const float* gptr,
                                            unsigned tx, unsigned ty,
                                            unsigned long long strideElems,
                                            unsigned d0flags) {
    unsigned long long ga = (unsigned long long)(uintptr_t)gptr;
    v4u g0;
    g0.x = 1u;                                         // count=1 (valid user D#)
    g0.y = ldsByte;                                    // lds_addr
    g0.z = (unsigned)ga;                               // global_addr[31:0]
    g0.w = ((unsigned)(ga >> 32) & 0x01FFFFFFu)        // global_addr[56:32]
           | 0x80000000u;                              // type=2 ("image")
    v8i g1;
    g1[0] = (int)d0flags;                              // mask=0, data_size=4B, pads
    g1[1] = (int)((tx & 0xFFFFu) << 16);               // tensor_dim0[15:0]
    g1[2] = (int)((tx >> 16) | ((ty & 0xFFFFu) << 16));// t_dim0[31:16]|t_dim1[15:0]
    g1[3] = (int)((ty >> 16) | (tx << 16));            // t_dim1[31:16]|tile_dim0
    g1[4] = (int)(ty & 0xFFFFu);                       // tile_dim1 (tile_dim2=0)
    g1[5] = (int)(strideElems & 0xFFFFFFFFull);        // dim0_stride[31:0]
    g1[6] = (int)((strideElems >> 32) & 0xFFFFull);    // dim0_stride[47:32]
    g1[7] = 0;                                         // dim1_stride (unused, 2-D)
    v4i zz = {};
#if __has_include(<hip/amd_detail/amd_gfx1250_TDM.h>)
    v8i z8 = {};
    __builtin_amdgcn_tensor_load_to_lds(g0, g1, zz, zz, z8, 0);
#else
    __builtin_amdgcn_tensor_load_to_lds(g0, g1, zz, zz, 0);
#endif
}

// ---------------------------------------------------------------- prep ------
__global__ __launch_bounds__(256)
void prep_kernel(const float* __restrict__ inp, const float* __restrict__ pos,
                 float* __restrict__ xbuf, float* __restrict__ kvbuf) {
    int i = blockIdx.x * 256 + threadIdx.x;
    int row = i >> 12;
    float v = 0.0f;
    if (row < S_) v = inp[i] + pos[i];
    xbuf[i]  = v;
    kvbuf[i] = v;
}

__global__ __launch_bounds__(256)
void zero_kernel(float4* __restrict__ p, int n4) {
    int i = blockIdx.x * 256 + threadIdx.x;
    if (i < n4) p[i] = make_float4(0.f, 0.f, 0.f, 0.f);
}

// ------------------------------------------- TDM + WMMA split-K GEMM --------
// headed=0: O[32,4096](ld D_)  += A[32,4096] * W[4096,4096]
// headed=1: O[h][32,512](ld DH_)+= A[32,4096] * W[h][4096][512]
__global__ __launch_bounds__(WAVES * 32)
void gemm_tdm_kernel(const float* __restrict__ A, const float* __restrict__ W,
                     float* __restrict__ O, int ldb, int headed) {
    __shared__ float smem[SMEM_FLOATS];

    const int lane = threadIdx.x & 31;
    const int wid  = threadIdx.x >> 5;
    const int hl   = lane >> 4;           // lane half: K pair / row+8 select
    const int mrow = lane & 15;
    const int ncol = lane & 15;
    const int colw = wid * 16;            // wave's 16-col slice inside 64-tile
    const int c0   = blockIdx.x * (WAVES * 16);

    const float* __restrict__ B;
    float* __restrict__ Out;
    int ldo, colbase;
    if (headed) {
        const int h = c0 >> 9;
        B = W + (size_t)h * D_ * DH_;
        Out = O + (size_t)h * (MPAD * DH_);
        ldo = DH_;  colbase = c0 & 511;
    } else {
        B = W;  Out = O;  ldo = D_;  colbase = c0;
    }

    const int klen = D_ / KSPLIT;
    const int k0   = blockIdx.y * klen;
    const int k1   = k0 + klen;

    // LDS byte offsets derived from the real shared address (escapes smem).
    const unsigned ldsBase = (unsigned)(size_t)(void*)smem;
    const unsigned offA0 = ldsBase;
    const unsigned offA1 = ldsBase + A_BUF_FLOATS * 4u;
    const unsigned offB0 = ldsBase + 2u * A_BUF_FLOATS * 4u;
    const unsigned offB1 = offB0 + B_BUF_FLOATS * 4u;

    v8f acc0 = {};
    v8f acc1 = {};

    // prologue: wave0 TDM-loads tile 0 into buffer 0
    const float* gA = A + k0;                              // rows 0..31, cols kt..
    const float* gB = B + (size_t)k0 * ldb + colbase;      // rows kt.., 64 cols
    if (wid == 0) {
        tdm_load_2d(offA0, gA, KT, 32, (unsigned long long)D_, TDM_D0_PADA);
        tdm_load_2d(offB0, gB, 64, KT, (unsigned long long)ldb, TDM_D0_PLAIN);
        __builtin_amdgcn_s_wait_tensorcnt(0);
    }
    __syncthreads();
    gA += KT;
    gB += (size_t)KT * ldb;

    int cur = 0;
    for (int kt = k0; kt < k1; kt += KT) {
        const int more = (kt + KT < k1);
        if (wid == 0 && more) {   // prefetch next tile into the other buffer
            tdm_load_2d(cur ? offA0 : offA1, gA, KT, 32,
                        (unsigned long long)D_, TDM_D0_PADA);
            tdm_load_2d(cur ? offB0 : offB1, gB, 64, KT,
                        (unsigned long long)ldb, TDM_D0_PLAIN);
        }

        lds_publish(smem);    // buffer `cur` is ready (waited+barrier'd earlier)
        const float* lAc = smem + (cur ? A_BUF_FLOATS : 0);
        const float* lBc = smem + 2 * A_BUF_FLOATS + (cur ? B_BUF_FLOATS : 0);

#pragma unroll
        for (int j = 0; j < KT; j += 4) {
            const int ka = j + 2 * hl;
            v2f a0 = *(const v2f*)(lAc + mrow * A_STRIDE + ka);         // rows 0..15
            v2f a1 = *(const v2f*)(lAc + (mrow + 16) * A_STRIDE + ka);  // rows 16..31
            v2f b;
            b.x = lBc[ka * 64 + colw + ncol];
            b.y = lBc[(ka + 1) * 64 + colw + ncol];
            acc0 = wmma4(a0, b, acc0);
            acc1 = wmma4(a1, b, acc1);
        }

        if (wid == 0 && more) __builtin_amdgcn_s_wait_tensorcnt(0);
        __syncthreads();
        gA += KT;
        gB += (size_t)KT * ldb;
        cur ^= 1;
    }

    // C/D layout: VGPR v, lanes 0-15 -> M=v, lanes 16-31 -> M=v+8
    const int col = colbase + colw + ncol;
#pragma unroll
    for (int v = 0; v < 8; ++v) {
        const int r = v + 8 * hl;
        atomicAdd(&Out[(size_t)r * ldo + col],        acc0[v]);
        atomicAdd(&Out[(size_t)(r + 16) * ldo + col], acc1[v]);
    }
}

// ------------------------------------------------------------- attention ----
__global__ __launch_bounds__(256)
void attn_kernel(const float* __restrict__ Q, const float* __restrict__ K,
                 const float* __restrict__ V,
                 const float* __restrict__ bq, const float* __restrict__ bk,
                 const float* __restrict__ bv, float* __restrict__ ctx) {
    const int h = blockIdx.x;
    __shared__ float sc[S_][S_];
    const float* __restrict__ Qh  = Q  + (size_t)h * MPAD * DH_;
    const float* __restrict__ Kh  = K  + (size_t)h * MPAD * DH_;
    const float* __restrict__ Vh  = V  + (size_t)h * MPAD * DH_;
    const float* __restrict__ bqh = bq + (size_t)h * DH_;
    const float* __restrict__ bkh = bk + (size_t)h * DH_;
    const float* __restrict__ bvh = bv + (size_t)h * DH_;

    for (int p = threadIdx.x; p < S_ * S_; p += 256) {
        const int s = p / S_, t = p % S_;
        float acc = 0.f;
        for (int e = 0; e < DH_; ++e)
            acc += (Qh[s * DH_ + e] + bqh[e]) * (Kh[t * DH_ + e] + bkh[e]);
        sc[s][t] = acc * INV_SCALE;
    }
    __syncthreads();

    if (threadIdx.x < S_) {
        const int s = threadIdx.x;
        float m = -1e30f;
        for (int t = 0; t < S_; ++t) m = fmaxf(m, sc[s][t]);
        float sum = 0.f;
        for (int t = 0; t < S_; ++t) { float ex = __expf(sc[s][t] - m); sc[s][t] = ex; sum += ex; }
        const float inv = 1.f / sum;
        for (int t = 0; t < S_; ++t) sc[s][t] *= inv;
    }
    __syncthreads();

    for (int p = threadIdx.x; p < S_ * DH_; p += 256) {
        const int s = p / DH_, e = p % DH_;
        float acc = 0.f;
        for (int t = 0; t < S_; ++t)
            acc += sc[s][t] * (Vh[t * DH_ + e] + bvh[e]);
        ctx[(size_t)s * D_ + h * DH_ + e] = acc;
    }
}

// ------------------------------------------------------------- layernorm ----
__global__ __launch_bounds__(256)
void ln_kernel(float* __restrict__ out, const float* __restrict__ a,
               const float* __restrict__ b, const float* __restrict__ bias) {
    const int row = blockIdx.x;
    __shared__ float red[256];
    if (row >= S_) {
        for (int j = threadIdx.x; j < D_; j += 256) out[(size_t)row * D_ + j] = 0.f;
        return;
    }
    float v[16];
    float lsum = 0.f;
#pragma unroll
    for (int i = 0; i < 16; ++i) {
        const int j = threadIdx.x + i * 256;
        float x = a[(size_t)row * D_ + j] + b[(size_t)row * D_ + j];
        if (bias) x += bias[j];
        v[i] = x;
        lsum += x;
    }
    red[threadIdx.x] = lsum; __syncthreads();
    for (int s = 128; s > 0; s >>= 1) {
        if (threadIdx.x < s) red[threadIdx.x] += red[threadIdx.x + s];
        __syncthreads();
    }
    const float mean = red[0] * (1.0f / D_);
    __syncthreads();
    float lvar = 0.f;
#pragma unroll
    for (int i = 0; i < 16; ++i) { const float d = v[i] - mean; lvar += d * d; }
    red[threadIdx.x] = lvar; __syncthreads();
    for (int s = 128; s > 0; s >>= 1) {
        if (threadIdx.x < s) red[threadIdx.x] += red[threadIdx.x + s];
        __syncthreads();
    }
    const float inv = rsqrtf(red[0] * (1.0f / D_) + EPS_);
#pragma unroll
    for (int i = 0; i < 16; ++i) {
        const int j = threadIdx.x + i * 256;
        out[(size_t)row * D_ + j] = (v[i] - mean) * inv;
    }
}

// ------------------------------------------------------------- epilogues ----
__global__ __launch_bounds__(256)
void bias_relu_kernel(float* __restrict__ t, const float* __restrict__ b1) {
    const int i = blockIdx.x * 256 + threadIdx.x;
    const int row = i >> 12, j = i & (D_ - 1);
    t[i] = (row < S_) ? fmaxf(t[i] + b1[j], 0.f) : 0.f;
}

__global__ __launch_bounds__(256)
void copyout_kernel(const float* __restrict__ xbuf, float* __restrict__ out) {
    const int i = blockIdx.x * 256 + threadIdx.x;
    out[i] = xbuf[i];
}

// ---------------------------------------------------------------------------
extern "C" void kernel_launch(void* const* d_in, const int* in_sizes, int n_in,
                              void* d_out, int out_size, void* d_ws, size_t ws_size,
                              hipStream_t stream) {
    const float* inp = (const float*)d_in[0];
    const float* pos = (const float*)d_in[1];
    const float* Wq  = (const float*)d_in[2];
    const float* bq  = (const float*)d_in[3];
    const float* Wk  = (const float*)d_in[4];
    const float* bk  = (const float*)d_in[5];
    const float* Wv  = (const float*)d_in[6];
    const float* bv  = (const float*)d_in[7];
    const float* W1  = (const float*)d_in[8];
    const float* b1  = (const float*)d_in[9];
    const float* W2  = (const float*)d_in[10];
    const float* b2  = (const float*)d_in[11];
    float* out = (float*)d_out;

    float* xbuf = (float*)d_ws;
    float* kv   = xbuf + ROWBUF;
    float* mh   = kv   + ROWBUF;
    float* ctx  = mh   + ROWBUF;
    float* t1   = ctx  + ROWBUF;   // t1..Vb zeroed together each layer
    float* t2   = t1   + ROWBUF;
    float* Qb   = t2   + ROWBUF;
    float* Kb   = Qb   + ROWBUF;
    float* Vb   = Kb   + ROWBUF;

    const size_t Wqkv_l = (size_t)H_ * D_ * DH_;
    const size_t bqkv_l = (size_t)H_ * DH_;
    const size_t Wff_l  = (size_t)D_ * D_;

    const dim3 gemm_grid(D_ / (WAVES * 16), KSPLIT);
    const int  gemm_blk = WAVES * 32;

    prep_kernel<<<ROWBUF / 256, 256, 0, stream>>>(inp, pos, xbuf, kv);

    for (int l = 0; l < L_; ++l) {
        zero_kernel<<<(5 * ROWBUF / 4 + 255) / 256, 256, 0, stream>>>(
            (float4*)t1, 5 * ROWBUF / 4);

        gemm_tdm_kernel<<<gemm_grid, gemm_blk, 0, stream>>>(
            xbuf, Wq + (size_t)l * Wqkv_l, Qb, DH_, 1);
        gemm_tdm_kernel<<<gemm_grid, gemm_blk, 0, stream>>>(
            kv,   Wk + (size_t)l * Wqkv_l, Kb, DH_, 1);
        gemm_tdm_kernel<<<gemm_grid, gemm_blk, 0, stream>>>(
            kv,   Wv + (size_t)l * Wqkv_l, Vb, DH_, 1);

        attn_kernel<<<H_, 256, 0, stream>>>(Qb, Kb, Vb,
                                            bq + (size_t)l * bqkv_l,
                                            bk + (size_t)l * bqkv_l,
                                            bv + (size_t)l * bqkv_l, ctx);

        ln_kernel<<<MPAD, 256, 0, stream>>>(mh, xbuf, ctx, nullptr);

        gemm_tdm_kernel<<<gemm_grid, gemm_blk, 0, stream>>>(
            mh, W1 + (size_t)l * Wff_l, t1, D_, 0);
        bias_relu_kernel<<<ROWBUF / 256, 256, 0, stream>>>(t1, b1 + (size_t)l * D_);
        gemm_tdm_kernel<<<gemm_grid, gemm_blk, 0, stream>>>(
            t1, W2 + (size_t)l * Wff_l, t2, D_, 0);

        ln_kernel<<<MPAD, 256, 0, stream>>>(xbuf, mh, t2, b2 + (size_t)l * D_);
    }

    copyout_kernel<<<(S_ * D_) / 256, 256, 0, stream>>>(xbuf, out);
}